// IndependentPatchEmbedding_14680198217848
// MI455X (gfx1250) — compile-verified
//
#include <hip/hip_runtime.h>
#include <stdint.h>

// IndependentPatchEmbedding: out[b,f,d] = sum_p x[b,f,p] * W[f,d,p] + bias[f,d]
//   x: [1024, 64, 128] f32, W: [64, 512, 128] f32, bias: [64, 512] f32
//   out: [1024, 64, 512] f32
//
// Per-feature GEMM, 64x64 output tile per workgroup, full K=128 staged ONCE
// into dynamic LDS (67,584 B; gfx1250 allows up to 320 KB/WG) via async
// global->LDS loads (ASYNCcnt path), exact-fp32 WMMA (v_wmma_f32_16x16x4_f32).
//
// LDS rows padded to 132 floats (528 B): 16B-aligned rows for b128 async
// stores, and row r starts at bank 4r -> a 16-row fragment read (ds_load_b64)
// touches all 64 banks exactly once. Fragment addresses are base_reg + ks*4B
// -> pure immediate-offset ds_loads, 2addr-fusible, no per-step VALU.

typedef __attribute__((ext_vector_type(2))) float v2f;
typedef __attribute__((ext_vector_type(8))) float v8f;

#define BTILE 64
#define DTILE 64
#define LDST  132                       // padded LDS row stride (floats)
#define LDS_BYTES ((BTILE + DTILE) * LDST * 4)   // 67,584 B (33 x 2048-B blocks)
#define NTHREADS 256

// Generic (shared) pointer -> 32-bit LDS byte address for async-load VDST.
__device__ __forceinline__ uint32_t lds_addr_of(const void* p) {
    return (uint32_t)(uintptr_t)(__attribute__((address_space(3))) const void*)p;
}

// GLOBAL_LOAD_ASYNC_TO_LDS_B128: each lane copies 16 B from the global address in vaddr to the LDS address in vdst.
// Tracked by ASYNCcnt; no VGPR data round-trip.
__device__ __forceinline__ void async_load_b128(uint32_t lds_byte_addr,
                                                const void* gaddr) {
    asm volatile("global_load_async_to_lds_b128 %0, %1, off"
                 :: "v"(lds_byte_addr), "v"(gaddr)
                 : "memory");
}

__device__ __forceinline__ void wait_asynccnt0() {
#if __has_builtin(__builtin_amdgcn_s_wait_asynccnt)
    __builtin_amdgcn_s_wait_asynccnt(0);
#else
    asm volatile("s_wait_asynccnt 0x0" ::: "memory");
#endif
}

__global__ __launch_bounds__(NTHREADS, 2)
void patch_embed_wmma_f32(const float* __restrict__ x,
                          const float* __restrict__ W,
                          const float* __restrict__ bias,
                          float* __restrict__ out)
{
    extern __shared__ float smem[];     // [BTILE*LDST] A tile | [DTILE*LDST] B tile
    float* As = smem;
    float* Bs = smem + BTILE * LDST;

    const int f  = blockIdx.z;              // feature 0..63
    const int b0 = blockIdx.y * BTILE;      // batch tile base
    const int d0 = blockIdx.x * DTILE;      // d tile base

    const int tid  = threadIdx.x;
    const int lane = tid & 31;
    const int wave = tid >> 5;              // 0..7

    // 8 waves: 4 M-tiles x 4 N-tiles; each wave owns one 16-row strip and
    // TWO adjacent 16-col tiles (shares the A fragment between 2 WMMAs).
    const int wm = wave & 3;                // M tile 0..3
    const int wn = (wave >> 2) << 1;        // N tile base: 0 or 2

    // ---- async-stage the whole 64x64xK=128 tile pair into LDS ----
    // 32 threads per row (float4 group g = 0..31), 8 rows per sweep, 8 sweeps.
    {
        const int g  = tid & 31;            // float4 group (column 4g)
        const int r0 = tid >> 5;            // 0..7
        const uint32_t dA = lds_addr_of(As) + (uint32_t)r0 * (LDST * 4u) + (uint32_t)g * 16u;
        const uint32_t dB = lds_addr_of(Bs) + (uint32_t)r0 * (LDST * 4u) + (uint32_t)g * 16u;
        const float* gA = x + ((size_t)(b0 + r0) * 64 + f) * 128 + 4 * g;
        const float* gB = W + ((size_t)f * 512 + (d0 + r0)) * 128 + 4 * g;
        #pragma unroll
        for (int p = 0; p < BTILE; p += 8) {
            async_load_b128(dA + (uint32_t)p * (LDST * 4u), gA + (size_t)p * 64 * 128);
            async_load_b128(dB + (uint32_t)p * (LDST * 4u), gB + (size_t)p * 128);
        }
    }
    wait_asynccnt0();       // this wave's LDS writes landed
    __syncthreads();        // everyone's LDS writes visible

    // ---- WMMA f32 16x16x4 fragment addressing (ISA 7.12.2):
    // lanes 0-15 hold K=k,k+1 at row=lane; lanes 16-31 hold K=k+2,k+3.
    const int row  = lane & 15;
    const int koff = (lane >> 4) << 1;      // 0 or 2

    const float* ap  = As + (wm * 16 + row) * LDST + koff;
    const float* bp0 = Bs + ((wn + 0) * 16 + row) * LDST + koff;
    const float* bp1 = Bs + ((wn + 1) * 16 + row) * LDST + koff;

    v8f acc0 = {};
    v8f acc1 = {};

    #pragma unroll
    for (int ks = 0; ks < 128; ks += 4) {
        v2f afrag  = *(const v2f*)(ap + ks);
        v2f bfrag0 = *(const v2f*)(bp0 + ks);
        v2f bfrag1 = *(const v2f*)(bp1 + ks);
        acc0 = __builtin_amdgcn_wmma_f32_16x16x4_f32(
            false, afrag, false, bfrag0, (short)0, acc0, false, false);
        acc1 = __builtin_amdgcn_wmma_f32_16x16x4_f32(
            false, afrag, false, bfrag1, (short)0, acc1, false, false);
    }

    // ---- epilogue: C/D layout VGPR r -> M=r (lanes 0-15) / r+8 (lanes 16-31),
    //      N = lane%16. Bias depends only on d. ----
    const int colL = lane & 15;
    const int c0 = d0 + (wn + 0) * 16 + colL;
    const int c1 = d0 + (wn + 1) * 16 + colL;
    const float bias0 = bias[f * 512 + c0];
    const float bias1 = bias[f * 512 + c1];
    const int rbase = b0 + wm * 16 + (lane >> 4) * 8;

    #pragma unroll
    for (int r = 0; r < 8; ++r) {
        size_t o = ((size_t)(rbase + r) * 64 + f) * 512;
        out[o + c0] = acc0[r] + bias0;
        out[o + c1] = acc1[r] + bias1;
    }
}

extern "C" void kernel_launch(void* const* d_in, const int* in_sizes, int n_in,
                              void* d_out, int out_size, void* d_ws, size_t ws_size,
                              hipStream_t stream) {
    (void)in_sizes; (void)n_in; (void)d_ws; (void)ws_size; (void)out_size;
    const float* x    = (const float*)d_in[0];   // [1024, 64, 128]
    const float* W    = (const float*)d_in[1];   // [64, 512, 128]
    const float* bias = (const float*)d_in[2];   // [64, 512]
    float* out        = (float*)d_out;           // [1024, 64, 512]

    dim3 grid(512 / DTILE, 1024 / BTILE, 64);    // (8, 16, 64) = 8192 workgroups
    dim3 block(NTHREADS);
    patch_embed_wmma_f32<<<grid, block, LDS_BYTES, stream>>>(x, W, bias, out);
}